// AttentionBase_89867895701943
// MI455X (gfx1250) — compile-verified
//
#include <hip/hip_runtime.h>
#include <hip/hip_bf16.h>

typedef __attribute__((ext_vector_type(2))) float v2f;
typedef __attribute__((ext_vector_type(8))) float v8f;
typedef unsigned int u32x4 __attribute__((ext_vector_type(4)));
typedef int          i32x8 __attribute__((ext_vector_type(8)));
typedef int          i32x4 __attribute__((ext_vector_type(4)));

#define HW    16384
#define C     192
#define BB    4
#define NH    8
#define HD    24
#define ROWS  (BB * C)      // 768
#define NT    32            // spatial columns per block in fused kernel
#define SEG   32            // K-split segments for the Gram-matrix kernel
#define SEGL  (HW / SEG)    // 512
#define KSTR  40            // LDS row stride (32 data + 8 TDM pad DWORDs)
#define EPSN  1e-12f

// ---------------------------------------------------------------------------
// Kernel 1: inverse L2 norms of the 768 q-rows and 768 k-rows (len 16384).
// ---------------------------------------------------------------------------
__global__ __launch_bounds__(256) void norm_kernel(const float* __restrict__ in1,
                                                   const float* __restrict__ in2,
                                                   float* __restrict__ invq,
                                                   float* __restrict__ invk) {
    __shared__ float red[256];
    int r = blockIdx.x;
    const float* src;
    float* dst;
    if (r < ROWS) { src = in1 + (size_t)r * HW;          dst = invq + r; }
    else          { src = in2 + (size_t)(r - ROWS) * HW; dst = invk + (r - ROWS); }
    float s = 0.f;
    const float4* p = (const float4*)src;
    for (int i = threadIdx.x; i < HW / 4; i += 256) {
        float4 v = p[i];
        s += v.x * v.x + v.y * v.y + v.z * v.z + v.w * v.w;
    }
    red[threadIdx.x] = s;
    __syncthreads();
    for (int off = 128; off > 0; off >>= 1) {
        if ((int)threadIdx.x < off) red[threadIdx.x] += red[threadIdx.x + off];
        __syncthreads();
    }
    if (threadIdx.x == 0) {
        float n = sqrtf(red[0]);
        *dst = 1.0f / fmaxf(n, EPSN);
    }
}

// ---------------------------------------------------------------------------
// Kernel 2a: raw Gram partials.  grid = (SEG, 32 bh), 4 waves/block; each wave
// owns one 16x16 WMMA tile of the padded 32x32 Gram matrix over K = SEGL.
// 1024 blocks keeps the 200 MB stream bandwidth-limited, not occupancy-limited.
// ---------------------------------------------------------------------------
__global__ __launch_bounds__(128) void attn_partial_kernel(const float* __restrict__ in1,
                                                           const float* __restrict__ in2,
                                                           float* __restrict__ partial) {
    int seg  = blockIdx.x;       // 0..SEG-1
    int bh   = blockIdx.y;       // 0..31
    int lane = threadIdx.x & 31;
    int wave = threadIdx.x >> 5; // 0..3
    int ti   = wave >> 1;
    int tj   = wave & 1;
    int lg   = lane >> 4;
    int ln   = lane & 15;

    int arow = bh * HD + ti * 16 + ln; if (arow > ROWS - 1) arow = ROWS - 1;
    int brow = bh * HD + tj * 16 + ln; if (brow > ROWS - 1) brow = ROWS - 1;
    const float* aptr = in1 + (size_t)arow * HW + seg * SEGL + 2 * lg;
    const float* bptr = in2 + (size_t)brow * HW + seg * SEGL + 2 * lg;

    v8f acc = {};
#pragma unroll 4
    for (int kb = 0; kb < SEGL; kb += 4) {
        v2f a = *(const v2f*)(aptr + kb);
        v2f b = *(const v2f*)(bptr + kb);
        acc = __builtin_amdgcn_wmma_f32_16x16x4_f32(false, a, false, b,
                                                    (short)0, acc, false, false);
    }
    float* po = partial + ((size_t)bh * SEG + seg) * 1024;
#pragma unroll
    for (int r = 0; r < 8; ++r) {
        int m = ti * 16 + r + 8 * lg;
        int n = tj * 16 + ln;
        po[m * 32 + n] = acc[r];
    }
}

// ---------------------------------------------------------------------------
// Kernel 2b: deterministic partial reduction + norm/scale fold + softmax.
// Also folds invk into the attention COLUMNS so the fused kernel can consume
// raw (unnormalized) k:  attn' = softmax(s*qn knT) * diag(invk).
// ---------------------------------------------------------------------------
__global__ __launch_bounds__(256) void attn_reduce_softmax(const float* __restrict__ partial,
                                                           const float* __restrict__ scale,
                                                           const float* __restrict__ invq,
                                                           const float* __restrict__ invk,
                                                           float* __restrict__ attn_out) {
    __shared__ float attnS[32][33];
    int bh = blockIdx.x;
    int h  = bh & 7;
    float sc = scale[h];
    for (int e = threadIdx.x; e < 1024; e += 256) {
        const float* p = partial + (size_t)bh * SEG * 1024 + e;
        float s = 0.f;
        for (int sg = 0; sg < SEG; ++sg) s += p[(size_t)sg * 1024];
        int m  = e >> 5;
        int n  = e & 31;
        int qi = bh * HD + (m < HD ? m : 0);   // padded region: value unused
        int ki = bh * HD + (n < HD ? n : 0);
        attnS[m][n] = s * invq[qi] * invk[ki] * sc;
    }
    __syncthreads();
    if (threadIdx.x < HD) {
        int row = threadIdx.x;
        float mx = -3.4e38f;
        for (int j = 0; j < HD; ++j) mx = fmaxf(mx, attnS[row][j]);
        float e[HD];
        float sum = 0.f;
        for (int j = 0; j < HD; ++j) { e[j] = __expf(attnS[row][j] - mx); sum += e[j]; }
        float inv = 1.0f / sum;
        float* o = attn_out + bh * (HD * HD) + row * HD;
        for (int j = 0; j < HD; ++j) o[j] = e[j] * inv * invk[bh * HD + j];
    }
}

// ---------------------------------------------------------------------------
// Kernel 3 (fused): per (batch, 32-column tile):
//   stage 0: TDM DMA of raw k tile (192 x 32 f32, row stride HW) into LDS,
//            with D# padding -> LDS row stride 40 DWORDs (bank spread)
//   stage 1: vT = per-head attn' @ kS   (WMMA; tmp never touches HBM)
//   stage 2: out tile = W @ vT          (WMMA)
// ---------------------------------------------------------------------------
__global__ __launch_bounds__(256) void fused_out_kernel(const float* __restrict__ in2,
                                                        const float* __restrict__ proj_w,
                                                        const float* __restrict__ attn,
                                                        float* __restrict__ out) {
    __shared__ float kS[C * KSTR];    // kS[c*KSTR + n], raw k tile (TDM target)
    __shared__ float vT[NT][C + 8];   // transposed tmp tile
    int bb   = blockIdx.y;
    int n0   = blockIdx.x * NT;
    int lane = threadIdx.x & 31;
    int wave = threadIdx.x >> 5;      // 0..7
    int lg   = lane >> 4;
    int ln   = lane & 15;

    // ---- stage 0: Tensor Data Mover load (wave 0 issues one DMA per block)
    if (threadIdx.x < 32) {
        unsigned long long gaddr = (unsigned long long)(uintptr_t)in2
                                 + ((unsigned long long)(bb * C) * HW + (unsigned)n0) * 4ull;
        unsigned int ldsa = (unsigned int)(uintptr_t)&kS[0];
        u32x4 g0;
        g0[0] = 1u;                                        // count=1 user descriptor
        g0[1] = ldsa;                                      // lds_addr
        g0[2] = (unsigned int)(gaddr & 0xFFFFFFFFu);       // global_addr[31:0]
        g0[3] = (unsigned int)((gaddr >> 32) & 0x1FFFFFFu) // global_addr[56:32]
              | 0x80000000u;                               // type=2 ("image")
        i32x8 g1;
        g1[0] = (int)(0x00020000u          // data_size = 2 -> 4 bytes
              |        0x00100000u         // pad_enable
              |        (4u << 22)          // pad_interval code 4 -> every 32 DWORDs
              |        (7u << 25));        // pad_amount  code 7 -> 8 DWORDs
        g1[1] = (int)((HW & 0xFFFFu) << 16);        // tensor_dim0[15:0]
        g1[2] = (int)((C  & 0xFFFFu) << 16);        // tensor_dim0[31:16]=0 | tensor_dim1[15:0]
        g1[3] = (int)((NT & 0xFFFFu) << 16);        // tensor_dim1[31:16]=0 | tile_dim0=32
        g1[4] = C;                                  // tile_dim1=192, tile_dim2=0 (unused)
        g1[5] = HW;                                 // tensor_dim0_stride[31:0]
        g1[6] = 0;                                  // stride hi / dim1_stride (2D: unused)
        g1[7] = 0;
        i32x4 z4 = {0, 0, 0, 0};                    // groups 2/3: zero (<=2D tensor)
        i32x8 z8 = {0, 0, 0, 0, 0, 0, 0, 0};
        __builtin_amdgcn_tensor_load_to_lds(g0, g1, z4, z4, z8, 0);
        __builtin_amdgcn_s_wait_tensorcnt(0);
    }
    __syncthreads();

    // ---- stage 1: jobs = head(8) x mtile(2) x ntile(2) = 32, 4 per wave
    for (int job = wave; job < 32; job += 8) {
        int hh = job >> 2;
        int mt = (job >> 1) & 1;
        int nt = job & 1;
        int m    = mt * 16 + ln;
        int meff = m < HD ? m : HD - 1;             // clamp padded rows (unused)
        const float* aptr = attn + (size_t)(bb * NH + hh) * (HD * HD) + meff * HD + 2 * lg;
        const float* kp   = kS + (size_t)(hh * HD + 2 * lg) * KSTR + (nt * 16 + ln);
        v8f acc = {};
#pragma unroll
        for (int kb = 0; kb < HD; kb += 4) {
            v2f a = *(const v2f*)(aptr + kb);
            v2f b = { kp[kb * KSTR], kp[kb * KSTR + KSTR] };
            acc = __builtin_amdgcn_wmma_f32_16x16x4_f32(false, a, false, b,
                                                        (short)0, acc, false, false);
        }
#pragma unroll
        for (int r = 0; r < 8; ++r) {
            int mm = mt * 16 + r + 8 * lg;
            if (mm < HD) vT[nt * 16 + ln][hh * HD + mm] = acc[r];
        }
    }
    __syncthreads();

    // ---- stage 2: out tile = W(192x192) @ vT(192x32); 24 jobs, 3 per wave
    for (int job = wave; job < 24; job += 8) {
        int mt = job >> 1;
        int nt = job & 1;
        int o  = mt * 16 + ln;
        const float* wptr = proj_w + (size_t)o * C + 2 * lg;
        const float* vp   = &vT[nt * 16 + ln][2 * lg];
        v8f acc = {};
#pragma unroll
        for (int kb = 0; kb < C; kb += 4) {
            v2f a = *(const v2f*)(wptr + kb);
            v2f b = *(const v2f*)(vp + kb);
            acc = __builtin_amdgcn_wmma_f32_16x16x4_f32(false, a, false, b,
                                                        (short)0, acc, false, false);
        }
        int ncol = n0 + nt * 16 + ln;
#pragma unroll
        for (int r = 0; r < 8; ++r) {
            int oo = mt * 16 + r + 8 * lg;
            out[(size_t)(bb * C + oo) * HW + ncol] = acc[r];
        }
    }
}

// ---------------------------------------------------------------------------
extern "C" void kernel_launch(void* const* d_in, const int* in_sizes, int n_in,
                              void* d_out, int out_size, void* d_ws, size_t ws_size,
                              hipStream_t stream) {
    const float* in1    = (const float*)d_in[0];
    const float* in2    = (const float*)d_in[1];
    const float* scale  = (const float*)d_in[2];
    const float* proj_w = (const float*)d_in[3];
    float* out = (float*)d_out;

    float* ws      = (float*)d_ws;
    float* invq    = ws;                         // 768
    float* invk    = ws + ROWS;                  // 768
    float* attn    = ws + 2 * ROWS;              // 32 * 576
    float* partial = attn + 32 * (HD * HD);      // 32 * SEG * 1024  (4 MB)

    norm_kernel<<<2 * ROWS, 256, 0, stream>>>(in1, in2, invq, invk);
    dim3 g2(SEG, BB * NH);
    attn_partial_kernel<<<g2, 128, 0, stream>>>(in1, in2, partial);
    attn_reduce_softmax<<<BB * NH, 256, 0, stream>>>(partial, scale, invq, invk, attn);
    dim3 g3(HW / NT, BB);
    fused_out_kernel<<<g3, 256, 0, stream>>>(in2, proj_w, attn, out);
}